// SimpleTransformer_7284264534116
// MI455X (gfx1250) — compile-verified
//
#include <hip/hip_runtime.h>
#include <math.h>

// ---------------------------------------------------------------------------
// CDNA5 (gfx1250) fp32 WMMA transformer.  wave32 only.
// ---------------------------------------------------------------------------

typedef __attribute__((ext_vector_type(2))) float v2f;
typedef __attribute__((ext_vector_type(8))) float v8f;

#define BDIM 128   // batch
#define SDIM 512   // src len
#define TDIM 64    // tgt len
#define DDIM 48    // model dim
#define FDIM 128   // ff dim
#define HDIM 4     // heads
#define HD   12    // head dim
#define SQRT_D 6.9282032302755088f
#define INV_SQRT_HD 0.28867513459481287f

// ---- WMMA f32 16x16x4 helpers (layouts per ISA 7.12.2) ---------------------

__device__ __forceinline__ v8f wmma4(v2f a, v2f b, v8f c) {
  return __builtin_amdgcn_wmma_f32_16x16x4_f32(false, a, false, b,
                                               (short)0, c, false, false);
}

// A fragment: 16 rows x 4 k.  lanes 0-15 hold k={kk,kk+1}, lanes 16-31 k={kk+2,kk+3}
__device__ __forceinline__ v2f frag_a(const float* A, int lda, int rbase, int kk) {
  int lane = threadIdx.x & 31;
  const float* p = A + (size_t)(rbase + (lane & 15)) * lda + kk + ((lane >> 4) << 1);
  v2f a; a.x = p[0]; a.y = p[1]; return a;
}

// B fragment: 4 k-rows x 16 cols of weight matrix W (K x N row-major, ld = ldw)
__device__ __forceinline__ v2f frag_b(const float* W, int ldw, int kk, int cbase) {
  int lane = threadIdx.x & 31;
  const float* p = W + (size_t)(kk + ((lane >> 4) << 1)) * ldw + cbase + (lane & 15);
  v2f b; b.x = p[0]; b.y = p[ldw]; return b;
}

// ---- generic GEMM: C[16x16 tile] = A(16xK) * W(KxN) + bias, optional relu --

__global__ void gemm_bias_kernel(const float* __restrict__ A, int lda,
                                 const float* __restrict__ W, int ldw,
                                 const float* __restrict__ bias,
                                 float* __restrict__ C, int ldc,
                                 int K, int relu) {
  int rbase = blockIdx.x * 16;
  int cbase = blockIdx.y * 16;
  v8f acc = {};
  for (int kk = 0; kk < K; kk += 4) {
    acc = wmma4(frag_a(A, lda, rbase, kk), frag_b(W, ldw, kk, cbase), acc);
  }
  int lane = threadIdx.x & 31;
  int half = lane >> 4;
  int col  = cbase + (lane & 15);
  float bv = bias[col];
  int row0 = rbase + half * 8;
#pragma unroll
  for (int r = 0; r < 8; ++r) {
    float v = acc[r] + bv;
    if (relu) v = fmaxf(v, 0.0f);
    C[(size_t)(row0 + r) * ldc + col] = v;
  }
}

// ---- fused GEMM (N=48) + bias + residual + LayerNorm (+ optional extra store
//      and fused out_w dot producing y) ---------------------------------------

__global__ void gemm_ln_kernel(const float* __restrict__ A, int lda, int K,
                               const float* __restrict__ W,      // K x 48
                               const float* __restrict__ bias,
                               const float* __restrict__ resid, int ldr,
                               const float* __restrict__ g,
                               const float* __restrict__ bta,
                               float* __restrict__ C, int ldc,
                               float* __restrict__ C2, int ldc2,
                               const float* __restrict__ outw,
                               const float* __restrict__ outb,
                               float* __restrict__ yout, int ystride) {
  int rbase = blockIdx.x * 16;
  v8f a0 = {}, a1 = {}, a2 = {};
  for (int kk = 0; kk < K; kk += 4) {
    v2f af = frag_a(A, lda, rbase, kk);
    a0 = wmma4(af, frag_b(W, 48, kk, 0),  a0);
    a1 = wmma4(af, frag_b(W, 48, kk, 16), a1);
    a2 = wmma4(af, frag_b(W, 48, kk, 32), a2);
  }
  int lane = threadIdx.x & 31;
  int half = lane >> 4;
  int c16  = lane & 15;
  float b0 = bias[c16], b1 = bias[16 + c16], b2 = bias[32 + c16];
  float g0 = g[c16],    g1 = g[16 + c16],   g2 = g[32 + c16];
  float t0 = bta[c16],  t1 = bta[16 + c16], t2 = bta[32 + c16];
  float w0 = 0.f, w1 = 0.f, w2 = 0.f;
  if (outw) { w0 = outw[c16]; w1 = outw[16 + c16]; w2 = outw[32 + c16]; }
#pragma unroll
  for (int r = 0; r < 8; ++r) {
    int row = rbase + half * 8 + r;
    const float* rp = resid + (size_t)row * ldr;
    float v0 = a0[r] + b0 + rp[c16];
    float v1 = a1[r] + b1 + rp[16 + c16];
    float v2 = a2[r] + b2 + rp[32 + c16];
    float s = v0 + v1 + v2;
    for (int m = 1; m < 16; m <<= 1) s += __shfl_xor(s, m, 16);
    float mean = s * (1.0f / 48.0f);
    float d0 = v0 - mean, d1 = v1 - mean, d2 = v2 - mean;
    float q = d0 * d0 + d1 * d1 + d2 * d2;
    for (int m = 1; m < 16; m <<= 1) q += __shfl_xor(q, m, 16);
    float rstd = rsqrtf(q * (1.0f / 48.0f) + 1e-5f);
    float o0 = d0 * rstd * g0 + t0;
    float o1 = d1 * rstd * g1 + t1;
    float o2 = d2 * rstd * g2 + t2;
    float* cp = C + (size_t)row * ldc;
    cp[c16] = o0; cp[16 + c16] = o1; cp[32 + c16] = o2;
    if (C2) {
      float* c2p = C2 + (size_t)row * ldc2;
      c2p[c16] = o0; c2p[16 + c16] = o1; c2p[32 + c16] = o2;
    }
    if (outw) {
      float y = o0 * w0 + o1 * w1 + o2 * w2;
      for (int m = 1; m < 16; m <<= 1) y += __shfl_xor(y, m, 16);
      if (c16 == 0) yout[(size_t)row * ystride] = y + outb[0];
    }
  }
}

// ---- encoder flash attention: one wave per (b, h, 16-query tile) -----------

__global__ void enc_attn_kernel(const float* __restrict__ Q,
                                const float* __restrict__ Kk,
                                const float* __restrict__ V,
                                float* __restrict__ O) {
  __shared__ float lds[16 * 17];
  int b = blockIdx.z, h = blockIdx.y, qt = blockIdx.x * 16;
  int lane = threadIdx.x & 31;
  int half = lane >> 4;
  int c16  = lane & 15;

  const float* qb = Q + ((size_t)(b * SDIM + qt)) * DDIM + h * HD;
  v2f aq[3];
#pragma unroll
  for (int i = 0; i < 3; ++i) {
    const float* p = qb + (size_t)c16 * DDIM + i * 4 + half * 2;
    aq[i].x = p[0]; aq[i].y = p[1];
  }
  float mrow[8], lrow[8];
  v8f oacc = {};
#pragma unroll
  for (int r = 0; r < 8; ++r) { mrow[r] = -1e30f; lrow[r] = 0.0f; }

  for (int key0 = 0; key0 < SDIM; key0 += 16) {
    // scores S(16q x 16k) = Q(16x12) * K^T(12x16)
    v8f s = {};
#pragma unroll
    for (int i = 0; i < 3; ++i) {
      const float* kp = Kk + ((size_t)(b * SDIM + key0 + c16)) * DDIM + h * HD
                        + i * 4 + half * 2;
      v2f bf; bf.x = kp[0]; bf.y = kp[1];
      s = wmma4(aq[i], bf, s);
    }
    // online softmax per row
    float p_[8];
#pragma unroll
    for (int r = 0; r < 8; ++r) {
      float sc = s[r] * INV_SQRT_HD;
      float mx = sc;
      for (int m = 1; m < 16; m <<= 1) mx = fmaxf(mx, __shfl_xor(mx, m, 16));
      float mnew = fmaxf(mrow[r], mx);
      float f = expf(mrow[r] - mnew);
      float p = expf(sc - mnew);
      float ps = p;
      for (int m = 1; m < 16; m <<= 1) ps += __shfl_xor(ps, m, 16);
      lrow[r] = lrow[r] * f + ps;
      mrow[r] = mnew;
      oacc[r] *= f;
      p_[r] = p;
    }
    __syncthreads();
#pragma unroll
    for (int r = 0; r < 8; ++r) lds[(half * 8 + r) * 17 + c16] = p_[r];
    __syncthreads();
    // O += P(16x16) * V(16k x 12)
#pragma unroll
    for (int kk = 0; kk < 16; kk += 4) {
      v2f ap;
      ap.x = lds[c16 * 17 + kk + half * 2];
      ap.y = lds[c16 * 17 + kk + half * 2 + 1];
      v2f bv;
      if (c16 < HD) {
        const float* vp = V + ((size_t)(b * SDIM + key0 + kk + half * 2)) * DDIM
                          + h * HD + c16;
        bv.x = vp[0]; bv.y = vp[DDIM];
      } else {
        bv.x = 0.0f; bv.y = 0.0f;
      }
      oacc = wmma4(ap, bv, oacc);
    }
    __syncthreads();
  }
  if (c16 < HD) {
#pragma unroll
    for (int r = 0; r < 8; ++r) {
      int row = qt + half * 8 + r;
      O[((size_t)(b * SDIM + row)) * DDIM + h * HD + c16] = oacc[r] / lrow[r];
    }
  }
}

// ---- decoder attention (GEMV style, one wave per (b,h)) --------------------

__global__ void dec_self_attn_kernel(const float* __restrict__ q,
                                     const float* __restrict__ Kc,
                                     const float* __restrict__ Vc,
                                     float* __restrict__ o, int t) {
  int b = blockIdx.x, h = blockIdx.y;
  int lane = threadIdx.x;
  float qv[HD];
#pragma unroll
  for (int i = 0; i < HD; ++i) qv[i] = q[b * DDIM + h * HD + i];
  int nk = t + 1;
  float sc[3];
  float mx = -1e30f;
#pragma unroll
  for (int j = 0; j < 3; ++j) {
    int key = lane + j * 32;
    float s = -1e30f;
    if (key < nk) {
      const float* kp = Kc + ((size_t)(b * (TDIM + 1) + key)) * DDIM + h * HD;
      float a = 0.f;
#pragma unroll
      for (int i = 0; i < HD; ++i) a += qv[i] * kp[i];
      s = a * INV_SQRT_HD;
    }
    sc[j] = s;
    mx = fmaxf(mx, s);
  }
  for (int m = 1; m < 32; m <<= 1) mx = fmaxf(mx, __shfl_xor(mx, m, 32));
  float lsum = 0.f;
  float oacc[HD];
#pragma unroll
  for (int i = 0; i < HD; ++i) oacc[i] = 0.f;
#pragma unroll
  for (int j = 0; j < 3; ++j) {
    int key = lane + j * 32;
    if (key < nk) {
      float p = expf(sc[j] - mx);
      lsum += p;
      const float* vp = Vc + ((size_t)(b * (TDIM + 1) + key)) * DDIM + h * HD;
#pragma unroll
      for (int i = 0; i < HD; ++i) oacc[i] += p * vp[i];
    }
  }
  for (int m = 1; m < 32; m <<= 1) lsum += __shfl_xor(lsum, m, 32);
#pragma unroll
  for (int i = 0; i < HD; ++i) {
    float v = oacc[i];
    for (int m = 1; m < 32; m <<= 1) v += __shfl_xor(v, m, 32);
    if (lane == 0) o[b * DDIM + h * HD + i] = v / lsum;
  }
}

__global__ void dec_cross_attn_kernel(const float* __restrict__ q,
                                      const float* __restrict__ Km,
                                      const float* __restrict__ Vm,
                                      float* __restrict__ o) {
  int b = blockIdx.x, h = blockIdx.y;
  int lane = threadIdx.x;
  float qv[HD];
#pragma unroll
  for (int i = 0; i < HD; ++i) qv[i] = q[b * DDIM + h * HD + i];
  float sc[16];
  float mx = -1e30f;
#pragma unroll
  for (int j = 0; j < 16; ++j) {
    int key = lane + j * 32;
    const float* kp = Km + ((size_t)(b * SDIM + key)) * DDIM + h * HD;
    float a = 0.f;
#pragma unroll
    for (int i = 0; i < HD; ++i) a += qv[i] * kp[i];
    sc[j] = a * INV_SQRT_HD;
    mx = fmaxf(mx, sc[j]);
  }
  for (int m = 1; m < 32; m <<= 1) mx = fmaxf(mx, __shfl_xor(mx, m, 32));
  float lsum = 0.f;
  float oacc[HD];
#pragma unroll
  for (int i = 0; i < HD; ++i) oacc[i] = 0.f;
#pragma unroll
  for (int j = 0; j < 16; ++j) {
    int key = lane + j * 32;
    float p = expf(sc[j] - mx);
    lsum += p;
    const float* vp = Vm + ((size_t)(b * SDIM + key)) * DDIM + h * HD;
#pragma unroll
    for (int i = 0; i < HD; ++i) oacc[i] += p * vp[i];
  }
  for (int m = 1; m < 32; m <<= 1) lsum += __shfl_xor(lsum, m, 32);
#pragma unroll
  for (int i = 0; i < HD; ++i) {
    float v = oacc[i];
    for (int m = 1; m < 32; m <<= 1) v += __shfl_xor(v, m, 32);
    if (lane == 0) o[b * DDIM + h * HD + i] = v / lsum;
  }
}

// ---- small elementwise kernels ---------------------------------------------

__global__ void pe_kernel(float* __restrict__ pe, int rows) {
  int i = blockIdx.x * blockDim.x + threadIdx.x;
  if (i < rows * 24) {
    int s = i / 24, k = i % 24;
    float div = expf((2.0f * k) * (-logf(10000.0f) / 48.0f));
    float a = (float)s * div;
    pe[s * DDIM + 2 * k]     = sinf(a);
    pe[s * DDIM + 2 * k + 1] = cosf(a);
  }
}

__global__ void embed_kernel(const float* __restrict__ src,
                             const float* __restrict__ in_w,
                             const float* __restrict__ in_b,
                             const float* __restrict__ pe,
                             float* __restrict__ x) {
  int i = blockIdx.x * blockDim.x + threadIdx.x;
  if (i < BDIM * SDIM * DDIM) {
    int rs = i / DDIM, j = i % DDIM;
    int s = rs % SDIM;
    x[i] = (src[rs] * in_w[j] + in_b[j]) * SQRT_D + pe[s * DDIM + j];
  }
}

__global__ void init_dec_kernel(const float* __restrict__ dec_start,
                                float* __restrict__ buf) {
  int i = blockIdx.x * blockDim.x + threadIdx.x;
  if (i < BDIM * DDIM) {
    int b = i / DDIM, j = i % DDIM;
    buf[(size_t)b * (TDIM + 1) * DDIM + j] = dec_start[j];
  }
}

__global__ void add_pe_kernel(const float* __restrict__ buf,
                              const float* __restrict__ pe,
                              float* __restrict__ qrow, int t) {
  int i = blockIdx.x * blockDim.x + threadIdx.x;
  if (i < BDIM * DDIM) {
    int b = i / DDIM, j = i % DDIM;
    qrow[i] = buf[((size_t)b * (TDIM + 1) + t) * DDIM + j] + pe[t * DDIM + j];
  }
}

// ---------------------------------------------------------------------------

extern "C" void kernel_launch(void* const* d_in, const int* in_sizes, int n_in,
                              void* d_out, int out_size, void* d_ws, size_t ws_size,
                              hipStream_t stream) {
  (void)in_sizes; (void)n_in; (void)out_size; (void)ws_size;
  const float* src       = (const float*)d_in[0];
  const float* in_w      = (const float*)d_in[1];
  const float* in_b      = (const float*)d_in[2];
  const float* dec_start = (const float*)d_in[3];
  const float* e_wq = (const float*)d_in[4];  const float* e_bq = (const float*)d_in[5];
  const float* e_wk = (const float*)d_in[6];  const float* e_bk = (const float*)d_in[7];
  const float* e_wv = (const float*)d_in[8];  const float* e_bv = (const float*)d_in[9];
  const float* e_wo = (const float*)d_in[10]; const float* e_bo = (const float*)d_in[11];
  const float* e_ln1_g = (const float*)d_in[12]; const float* e_ln1_b = (const float*)d_in[13];
  const float* e_w1 = (const float*)d_in[14]; const float* e_b1 = (const float*)d_in[15];
  const float* e_w2 = (const float*)d_in[16]; const float* e_b2 = (const float*)d_in[17];
  const float* e_ln2_g = (const float*)d_in[18]; const float* e_ln2_b = (const float*)d_in[19];
  const float* d_swq = (const float*)d_in[20]; const float* d_sbq = (const float*)d_in[21];
  const float* d_swk = (const float*)d_in[22]; const float* d_sbk = (const float*)d_in[23];
  const float* d_swv = (const float*)d_in[24]; const float* d_sbv = (const float*)d_in[25];
  const float* d_swo = (const float*)d_in[26]; const float* d_sbo = (const float*)d_in[27];
  const float* d_ln1_g = (const float*)d_in[28]; const float* d_ln1_b = (const float*)d_in[29];
  const float* d_cwq = (const float*)d_in[30]; const float* d_cbq = (const float*)d_in[31];
  const float* d_cwk = (const float*)d_in[32]; const float* d_cbk = (const float*)d_in[33];
  const float* d_cwv = (const float*)d_in[34]; const float* d_cbv = (const float*)d_in[35];
  const float* d_cwo = (const float*)d_in[36]; const float* d_cbo = (const float*)d_in[37];
  const float* d_ln2_g = (const float*)d_in[38]; const float* d_ln2_b = (const float*)d_in[39];
  const float* d_w1 = (const float*)d_in[40]; const float* d_b1 = (const float*)d_in[41];
  const float* d_w2 = (const float*)d_in[42]; const float* d_b2 = (const float*)d_in[43];
  const float* d_ln3_g = (const float*)d_in[44]; const float* d_ln3_b = (const float*)d_in[45];
  const float* out_w = (const float*)d_in[46]; const float* out_b = (const float*)d_in[47];
  float* yout = (float*)d_out;  // (B, T) row-major

  // workspace layout (floats)
  float* W = (float*)d_ws;
  const size_t NB = (size_t)BDIM * SDIM * DDIM;  // 3,145,728
  float* pe  = W;
  float* x   = W + 32768;
  float* Qe  = x  + NB;
  float* Ke  = Qe + NB;
  float* Ve  = Ke + NB;
  float* ao  = Ve + NB;
  float* x1  = ao + NB;
  float* hff = Qe;          // reuse (Q/K/V dead by then), needs 2.67*NB
  float* mem = x;           // reuse (x dead after attn-out LN)
  float* Km  = ao;          // reuse
  float* Vm  = x1;          // reuse (x1 dead after encoder LN2)
  float* buf  = x1 + NB;                       // (B, T+1, 48)
  float* Kc   = buf  + (size_t)BDIM * (TDIM + 1) * DDIM;
  float* Vc   = Kc   + (size_t)BDIM * (TDIM + 1) * DDIM;
  float* qrow = Vc   + (size_t)BDIM * (TDIM + 1) * DDIM;
  float* qs   = qrow + BDIM * DDIM;
  float* osf  = qs   + BDIM * DDIM;
  float* h1   = osf  + BDIM * DDIM;
  float* qc   = h1   + BDIM * DDIM;
  float* ocr  = qc   + BDIM * DDIM;
  float* h2   = ocr  + BDIM * DDIM;
  float* hfd  = h2   + BDIM * DDIM;            // (B, 128)
  float* h3   = hfd  + BDIM * FDIM;

  const int MROWS = BDIM * SDIM;               // 65536
  const int MT    = MROWS / 16;                // 4096 row tiles
  const int ldKV  = (TDIM + 1) * DDIM;         // decoder cache row stride

  // ---- encoder ----
  pe_kernel<<<(522 * 24 + 255) / 256, 256, 0, stream>>>(pe, 522);
  embed_kernel<<<(MROWS * DDIM + 255) / 256, 256, 0, stream>>>(src, in_w, in_b, pe, x);

  gemm_bias_kernel<<<dim3(MT, 3), 32, 0, stream>>>(x, DDIM, e_wq, DDIM, e_bq, Qe, DDIM, DDIM, 0);
  gemm_bias_kernel<<<dim3(MT, 3), 32, 0, stream>>>(x, DDIM, e_wk, DDIM, e_bk, Ke, DDIM, DDIM, 0);
  gemm_bias_kernel<<<dim3(MT, 3), 32, 0, stream>>>(x, DDIM, e_wv, DDIM, e_bv, Ve, DDIM, DDIM, 0);

  enc_attn_kernel<<<dim3(SDIM / 16, HDIM, BDIM), 32, 0, stream>>>(Qe, Ke, Ve, ao);

  gemm_ln_kernel<<<MT, 32, 0, stream>>>(ao, DDIM, DDIM, e_wo, e_bo, x, DDIM,
                                        e_ln1_g, e_ln1_b, x1, DDIM,
                                        nullptr, 0, nullptr, nullptr, nullptr, 0);
  gemm_bias_kernel<<<dim3(MT, 8), 32, 0, stream>>>(x1, DDIM, e_w1, FDIM, e_b1, hff, FDIM, DDIM, 1);
  gemm_ln_kernel<<<MT, 32, 0, stream>>>(hff, FDIM, FDIM, e_w2, e_b2, x1, DDIM,
                                        e_ln2_g, e_ln2_b, mem, DDIM,
                                        nullptr, 0, nullptr, nullptr, nullptr, 0);

  // precompute cross-attention K/V of mem (step-invariant)
  gemm_bias_kernel<<<dim3(MT, 3), 32, 0, stream>>>(mem, DDIM, d_cwk, DDIM, d_cbk, Km, DDIM, DDIM, 0);
  gemm_bias_kernel<<<dim3(MT, 3), 32, 0, stream>>>(mem, DDIM, d_cwv, DDIM, d_cbv, Vm, DDIM, DDIM, 0);

  // ---- decoder (KV-cached, one query row per step; exact vs reference) ----
  init_dec_kernel<<<(BDIM * DDIM + 255) / 256, 256, 0, stream>>>(dec_start, buf);

  const int DT = BDIM / 16;  // 8 row tiles
  for (int t = 0; t < TDIM; ++t) {
    add_pe_kernel<<<(BDIM * DDIM + 255) / 256, 256, 0, stream>>>(buf, pe, qrow, t);
    // self-attention projections; K/V appended to cache at position t
    gemm_bias_kernel<<<dim3(DT, 3), 32, 0, stream>>>(qrow, DDIM, d_swq, DDIM, d_sbq, qs, DDIM, DDIM, 0);
    gemm_bias_kernel<<<dim3(DT, 3), 32, 0, stream>>>(qrow, DDIM, d_swk, DDIM, d_sbk, Kc + t * DDIM, ldKV, DDIM, 0);
    gemm_bias_kernel<<<dim3(DT, 3), 32, 0, stream>>>(qrow, DDIM, d_swv, DDIM, d_sbv, Vc + t * DDIM, ldKV, DDIM, 0);
    dec_self_attn_kernel<<<dim3(BDIM, HDIM), 32, 0, stream>>>(qs, Kc, Vc, osf, t);
    gemm_ln_kernel<<<DT, 32, 0, stream>>>(osf, DDIM, DDIM, d_swo, d_sbo, qrow, DDIM,
                                          d_ln1_g, d_ln1_b, h1, DDIM,
                                          nullptr, 0, nullptr, nullptr, nullptr, 0);
    // cross attention
    gemm_bias_kernel<<<dim3(DT, 3), 32, 0, stream>>>(h1, DDIM, d_cwq, DDIM, d_cbq, qc, DDIM, DDIM, 0);
    dec_cross_attn_kernel<<<dim3(BDIM, HDIM), 32, 0, stream>>>(qc, Km, Vm, ocr);
    gemm_ln_kernel<<<DT, 32, 0, stream>>>(ocr, DDIM, DDIM, d_cwo, d_cbo, h1, DDIM,
                                          d_ln2_g, d_ln2_b, h2, DDIM,
                                          nullptr, 0, nullptr, nullptr, nullptr, 0);
    // FFN + final LN, fused with buffer append and y = h3 @ out_w + out_b
    gemm_bias_kernel<<<dim3(DT, 8), 32, 0, stream>>>(h2, DDIM, d_w1, FDIM, d_b1, hfd, FDIM, DDIM, 1);
    gemm_ln_kernel<<<DT, 32, 0, stream>>>(hfd, FDIM, FDIM, d_w2, d_b2, h2, DDIM,
                                          d_ln3_g, d_ln3_b, h3, DDIM,
                                          buf + (size_t)(t + 1) * DDIM, ldKV,
                                          out_w, out_b, yout + t, TDIM);
  }
}